// LayerNodeAttention_59373627900191
// MI455X (gfx1250) — compile-verified
//
#include <hip/hip_runtime.h>
#include <hip/hip_bf16.h>
#include <math.h>

// GAT-style layer for MI455X (gfx1250, wave32).
// Main GEMM: V_WMMA_F32_16X16X4_F32 (fp32 in/out, bit-faithful to fp32 ref).
// One workgroup per node n; 4 waves; wave w owns heads {2w, 2w+1} as the 16
// columns of one WMMA D tile. Epilogue: softmax rows distributed one-per-lane,
// all cross-lane traffic via single-instruction ds_swizzle_b32 patterns.

typedef __attribute__((ext_vector_type(2))) float v2f;
typedef __attribute__((ext_vector_type(4))) float v4f;
typedef __attribute__((ext_vector_type(8))) float v8f;

#define LRELU_ALPHA 0.2f
constexpr int Mn = 8;     // rows per node
constexpr int Nn = 4096;  // nodes
constexpr int Fn = 256;   // features
constexpr int Hn = 8;     // heads
constexpr int On = 8;     // out per head
constexpr int LDA = 268;  // padded LDS row stride (multiple of 4, bank-spread)

// ds_swizzle_b32 on a float; offset must be a macro-pasted constant.
#define SWZ(v, off) \
  __int_as_float(__builtin_amdgcn_ds_swizzle(__float_as_int(v), (off)))

// One xor-butterfly round over both partial vectors (masks < 8 keep sources
// inside the 8-lane head group).
#define RED_ROUND(p1, p2, d)                              \
  _Pragma("unroll") for (int m = 0; m < 8; ++m) {         \
    p1[m] += SWZ(p1[m], ((d) << 10) | 0x1f);              \
    p2[m] += SWZ(p2[m], ((d) << 10) | 0x1f);              \
  }

// out[mm] = sum_k attn[mm][k] * x[k]; attn row mm lives on lane gb+mm
// (that lane's own softmax row). Broadcast: new_lane = (lane & 0x18) | mm.
#define ATT_STEP(mm)                                      \
  {                                                       \
    float om = 0.f;                                       \
    _Pragma("unroll") for (int k = 0; k < 8; ++k)         \
        om += SWZ(at[k], ((mm) << 5) | 0x18) * x[k];      \
    out[mm] = om;                                         \
  }

// gat_block for one head-group of 8 lanes. Lane owns column o = oo of x
// (x[m] = column values); also computes softmax row m = oo for the group.
__device__ __forceinline__ void gat_block8(const float* __restrict__ x,
                                           float a1, float a2, int oo,
                                           float* __restrict__ out) {
  float p1[8], p2[8];
#pragma unroll
  for (int m = 0; m < 8; ++m) { p1[m] = x[m] * a1; p2[m] = x[m] * a2; }
  RED_ROUND(p1, p2, 1)
  RED_ROUND(p1, p2, 2)
  RED_ROUND(p1, p2, 4)
  // Full e1[.] / e2[.] now at every lane. Select this lane's row value e1[oo].
  float e1r = p1[0];
#pragma unroll
  for (int m = 1; m < 8; ++m) e1r = (oo == m) ? p1[m] : e1r;

  // Own-row softmax: attn[oo][k], k = 0..7.
  float at[8];
  float mx = -INFINITY;
#pragma unroll
  for (int k = 0; k < 8; ++k) {
    float v = e1r + p2[k];
    v = (v >= 0.f) ? v : LRELU_ALPHA * v;   // leaky_relu
    at[k] = v;
    mx = fmaxf(mx, v);
  }
  float den = 0.f;
#pragma unroll
  for (int k = 0; k < 8; ++k) {
    const float e = __expf(at[k] - mx);
    at[k] = e;
    den += e;
  }
  const float rden = __builtin_amdgcn_rcpf(den);
#pragma unroll
  for (int k = 0; k < 8; ++k) at[k] *= rden;

  // Combine: gather each row's attn from its owner lane, apply to own column.
  ATT_STEP(0) ATT_STEP(1) ATT_STEP(2) ATT_STEP(3)
  ATT_STEP(4) ATT_STEP(5) ATT_STEP(6) ATT_STEP(7)
}

// x2[m] += Z_col_k[m] * t3[k][o]; Z column k via in-group broadcast swizzle.
#define T3_STEP(k)                                        \
  {                                                       \
    const float t3v = t3row[(size_t)(k) * On];            \
    _Pragma("unroll") for (int m = 0; m < 8; ++m)         \
        x2[m] += SWZ(z[m], ((k) << 5) | 0x18) * t3v;      \
  }

__global__ __launch_bounds__(128) void gat_layer_kernel(
    const float* __restrict__ nf,   // node_features (M, N, F)
    const float* __restrict__ t1,   // trans1 (N, H, F, o)
    const float* __restrict__ t3,   // trans3 (N, H, M, o)
    const float* __restrict__ att,  // att (N, H, 2o)
    float* __restrict__ Y) {        // out (M, N, H*o)
  __shared__ float lA[16 * LDA];

  const int n    = blockIdx.x;
  const int tid  = threadIdx.x;
  const int lane = tid & 31;   // wave32 lane
  const int w    = tid >> 5;   // wave id 0..3

  // ---- Stage A = laa[n] (8x256) into LDS; zero rows 8..15 for the WMMA tile.
  for (int i = tid; i < 8 * LDA; i += 128) lA[8 * LDA + i] = 0.f;
  for (int i = tid; i < (Mn * Fn) / 4; i += 128) {
    const int m = i >> 6;             // 64 float4 per 256-float row
    const int f = (i & 63) << 2;
    const v4f v = __builtin_nontemporal_load(
        (const v4f*)(nf + (size_t)m * Nn * Fn + (size_t)n * Fn + f));
    *(v4f*)(&lA[m * LDA + f]) = v;
  }
  __syncthreads();

  // ---- Main GEMM: D(16x16) += A(16x4) * B(4x16), K = 256 in steps of 4.
  // A layout (ISA 16x4 f32): lanes 0-15 -> M=lane, K0/K1; lanes 16-31 -> K2/K3.
  // B layout mirrored: lane%16 = column N, same K split.
  const int m16  = lane & 15;
  const int koff = (lane >> 4) << 1;           // 0 or 2
  const int oo   = lane & 7;                   // in-group index / output o
  const int h    = 2 * w + ((lane >> 3) & 1);  // head for this lane's column
  const float* bptr = t1 + (((size_t)n * Hn + h) * Fn) * On + oo;

  v8f c = {0.f, 0.f, 0.f, 0.f, 0.f, 0.f, 0.f, 0.f};
#pragma unroll 8
  for (int k0 = 0; k0 < Fn; k0 += 4) {
    const int fa = k0 + koff;
    v2f a, b;
    a.x = lA[m16 * LDA + fa];
    a.y = lA[m16 * LDA + fa + 1];
    b.x = __builtin_nontemporal_load(bptr + (size_t)fa * On);        // streamed once
    b.y = __builtin_nontemporal_load(bptr + (size_t)(fa + 1) * On);
    c = __builtin_amdgcn_wmma_f32_16x16x4_f32(false, a, false, b,
                                              (short)0, c, false, false);
  }

  // ---- Epilogue: lanes 0-15 hold valid D columns (M=0..7 in c[0..7]).
  if (lane < 16) {
    float xc[8];
#pragma unroll
    for (int g = 0; g < 8; ++g) xc[g] = c[g];

    const size_t nh = (size_t)n * Hn + h;
    const float a1 = att[nh * (2 * On) + oo];
    const float a2 = att[nh * (2 * On) + On + oo];
    const float* __restrict__ t3row = t3 + nh * (Mn * On) + oo;

    float z[8];
    gat_block8(xc, a1, a2, oo, z);        // Z column o

    // x2[m][o] = sum_k Z[m][k] * trans3[n][h][k][o]
    float x2[8] = {0, 0, 0, 0, 0, 0, 0, 0};
    T3_STEP(0) T3_STEP(1) T3_STEP(2) T3_STEP(3)
    T3_STEP(4) T3_STEP(5) T3_STEP(6) T3_STEP(7)

    float o2[8];
    gat_block8(x2, a1, a2, oo, o2);       // second gat block

    // Y[m][n][h*8 + o] = sigmoid(out2[m][o])
#pragma unroll
    for (int m = 0; m < 8; ++m) {
      const float y = __builtin_amdgcn_rcpf(1.f + __expf(-o2[m]));
      __builtin_nontemporal_store(
          y, Y + ((size_t)m * Nn + n) * (Hn * On) + h * On + oo);
    }
  }
}

extern "C" void kernel_launch(void* const* d_in, const int* in_sizes, int n_in,
                              void* d_out, int out_size, void* d_ws, size_t ws_size,
                              hipStream_t stream) {
  (void)in_sizes; (void)n_in; (void)out_size; (void)d_ws; (void)ws_size;
  const float* nf  = (const float*)d_in[0];  // node_features (8, 4096, 256)
  const float* t1  = (const float*)d_in[1];  // trans1 (4096, 8, 256, 8)
  const float* t3  = (const float*)d_in[2];  // trans3 (4096, 8, 8, 8)
  const float* att = (const float*)d_in[3];  // att (4096, 8, 16)
  float* Y = (float*)d_out;                  // (8, 4096, 64)

  hipLaunchKernelGGL(gat_layer_kernel, dim3(Nn), dim3(128), 0, stream,
                     nf, t1, t3, att, Y);
}